// VQCBottleneck_84748294685002
// MI455X (gfx1250) — compile-verified
//
#include <hip/hip_runtime.h>
#include <math.h>

typedef __attribute__((ext_vector_type(2))) float v2f;
typedef __attribute__((ext_vector_type(8))) float v8f;

#define B_   64
#define S_   512
#define VD_  1024
#define NQ_  4
#define LD_  512
#define NL_  2
#define SPLIT_ 8
#define SCHUNK_ (S_ / SPLIT_)   // 64

// ---------------------------------------------------------------------------
// K1: partial sums over S. grid (64, 8), block 256. Each thread owns one
// float4 (4 consecutive v) and sums 64 s-slices. Fully coalesced 128B lines.
// This kernel is the roofline term: 134 MB @ 23.3 TB/s ~= 5.7 us.
// ---------------------------------------------------------------------------
__global__ void k_partial(const float* __restrict__ x, float* __restrict__ P) {
    int b = blockIdx.x;          // batch
    int p = blockIdx.y;          // s-chunk
    int t = threadIdx.x;         // float4 index over VD (256 * 4 = 1024)
    const float4* xb = (const float4*)(x + (size_t)b * S_ * VD_);
    float4 acc; acc.x = 0.f; acc.y = 0.f; acc.z = 0.f; acc.w = 0.f;
    int s0 = p * SCHUNK_;
    #pragma unroll 4
    for (int s = 0; s < SCHUNK_; ++s) {
        float4 v = xb[(size_t)(s0 + s) * (VD_ / 4) + t];
        acc.x += v.x; acc.y += v.y; acc.z += v.z; acc.w += v.w;
    }
    ((float4*)(P + ((size_t)p * B_ + b) * VD_))[t] = acc;
}

// ---------------------------------------------------------------------------
// K2: reduce 8 partials -> xm (with 1/S scale). grid 64, block 256.
// ---------------------------------------------------------------------------
__global__ void k_reduce(const float* __restrict__ P, float* __restrict__ XM) {
    int idx = blockIdx.x * blockDim.x + threadIdx.x;   // over 64*256 float4s
    float4 acc; acc.x = 0.f; acc.y = 0.f; acc.z = 0.f; acc.w = 0.f;
    #pragma unroll
    for (int p = 0; p < SPLIT_; ++p) {
        float4 v = ((const float4*)P)[(size_t)p * (B_ * VD_ / 4) + idx];
        acc.x += v.x; acc.y += v.y; acc.z += v.z; acc.w += v.w;
    }
    const float inv = 1.0f / (float)S_;
    acc.x *= inv; acc.y *= inv; acc.z *= inv; acc.w *= inv;
    ((float4*)XM)[idx] = acc;
}

// ---------------------------------------------------------------------------
// K3: z = tanh(xm @ W_red^T + b_red) * pi   via V_WMMA_F32_16X16X4_F32.
// M=64 -> 4 waves (one 16-row tile each), N=4 padded to 16, K=1024 in
// 256 steps of K=4. 1 block of 128 threads (4 waves).
// A layout: lanes 0-15: K=k,k+1 ; lanes 16-31: K=k+2,k+3 (per ISA 16x4 f32).
// B layout: vgpr0/1 = rows K0/K1 (lanes 0-15), K2/K3 (lanes 16-31).
// Zero-padding of B columns 4..15 is done branchlessly (clamped row + 0.0f
// multiplier) so the hot loop has no EXEC-mask save/restore around loads.
// ---------------------------------------------------------------------------
__global__ void k_z(const float* __restrict__ XM, const float* __restrict__ W_red,
                    const float* __restrict__ b_red, float* __restrict__ Z) {
    int lane = threadIdx.x & 31;
    int mt   = threadIdx.x >> 5;        // M tile 0..3
    int n16  = lane & 15;
    bool hi  = lane >= 16;
    int koff = hi ? 2 : 0;

    const float* arow = XM + (size_t)(mt * 16 + n16) * VD_ + koff;
    int   nr    = (n16 < NQ_) ? n16 : (NQ_ - 1);     // clamped, always in-bounds
    float wmask = (n16 < NQ_) ? 1.0f : 0.0f;         // zero-pad columns 4..15
    const float* brow = W_red + (size_t)nr * VD_ + koff;

    v8f acc = {0.f, 0.f, 0.f, 0.f, 0.f, 0.f, 0.f, 0.f};
    #pragma unroll 8
    for (int k = 0; k < VD_; k += 4) {
        v2f a, bm;
        a.x  = arow[k];
        a.y  = arow[k + 1];
        bm.x = brow[k] * wmask;
        bm.y = brow[k + 1] * wmask;
        acc = __builtin_amdgcn_wmma_f32_16x16x4_f32(
            false, a, false, bm, (short)0, acc, false, false);
    }
    // D: vgpr r -> row mt*16 + r (+8 for hi lanes), col n16
    if (n16 < NQ_) {
        float bias = b_red[n16];
        #pragma unroll
        for (int r = 0; r < 8; ++r) {
            int row = mt * 16 + r + (hi ? 8 : 0);
            Z[row * NQ_ + n16] = tanhf(acc[r] + bias) * 3.14159274101257324f;
        }
    }
}

// ---------------------------------------------------------------------------
// K4a: 4-qubit statevector simulation, one batch element per lane.
// 1 block of 64 threads; state (16 complex amplitudes) fully unrolled so it
// stays in VGPRs. wire w <-> bit (3-w); CNOT(c, c+1); PauliZ expvals.
// ---------------------------------------------------------------------------
__global__ void k_sim(const float* __restrict__ Z, const float* __restrict__ W,
                      float* __restrict__ Q) {
    int b = threadIdx.x;
    float re[16], im[16];
    #pragma unroll
    for (int i = 0; i < 16; ++i) { re[i] = 0.f; im[i] = 0.f; }
    re[0] = 1.f;

    // Initial RY(z[w]) on each wire (real 2x2 [[c,-s],[s,c]])
    #pragma unroll
    for (int w = 0; w < NQ_; ++w) {
        float ang = Z[b * NQ_ + w] * 0.5f;
        float s, c; sincosf(ang, &s, &c);
        int mask = 1 << (3 - w);
        #pragma unroll
        for (int i0 = 0; i0 < 16; ++i0) {
            if (i0 & mask) continue;
            int i1 = i0 | mask;
            float r0 = re[i0], m0 = im[i0], r1 = re[i1], m1 = im[i1];
            re[i0] = c * r0 - s * r1;  im[i0] = c * m0 - s * m1;
            re[i1] = s * r0 + c * r1;  im[i1] = s * m0 + c * m1;
        }
    }

    // Layers: Rot(phi,theta,omega) = RZ(omega) RY(theta) RZ(phi), then CNOTs
    #pragma unroll
    for (int l = 0; l < NL_; ++l) {
        #pragma unroll
        for (int w = 0; w < NQ_; ++w) {
            float phi   = W[((l * NQ_ + w) * 3) + 0];
            float theta = W[((l * NQ_ + w) * 3) + 1];
            float omega = W[((l * NQ_ + w) * 3) + 2];
            float st, ct; sincosf(theta * 0.5f, &st, &ct);
            float sp, cp; sincosf((phi + omega) * 0.5f, &sp, &cp);
            float sm, cm; sincosf((phi - omega) * 0.5f, &sm, &cm);
            // U00 = e^{-i(phi+omega)/2} ct ; U01 = -e^{+i(phi-omega)/2} st
            // U10 = e^{-i(phi-omega)/2} st ; U11 = e^{+i(phi+omega)/2} ct
            float u00r =  ct * cp, u00i = -ct * sp;
            float u01r = -st * cm, u01i = -st * sm;
            float u10r =  st * cm, u10i = -st * sm;
            float u11r =  ct * cp, u11i =  ct * sp;
            int mask = 1 << (3 - w);
            #pragma unroll
            for (int i0 = 0; i0 < 16; ++i0) {
                if (i0 & mask) continue;
                int i1 = i0 | mask;
                float r0 = re[i0], m0 = im[i0], r1 = re[i1], m1 = im[i1];
                re[i0] = u00r * r0 - u00i * m0 + u01r * r1 - u01i * m1;
                im[i0] = u00r * m0 + u00i * r0 + u01r * m1 + u01i * r1;
                re[i1] = u10r * r0 - u10i * m0 + u11r * r1 - u11i * m1;
                im[i1] = u10r * m0 + u10i * r0 + u11r * m1 + u11i * r1;
            }
        }
        #pragma unroll
        for (int c = 0; c < NQ_ - 1; ++c) {
            int cmask = 1 << (3 - c);
            int tmask = 1 << (2 - c);
            #pragma unroll
            for (int i = 0; i < 16; ++i) {
                if ((i & cmask) && !(i & tmask)) {
                    int j = i | tmask;
                    float tr = re[i]; re[i] = re[j]; re[j] = tr;
                    float ti = im[i]; im[i] = im[j]; im[j] = ti;
                }
            }
        }
    }

    // PauliZ expectation values
    #pragma unroll
    for (int w = 0; w < NQ_; ++w) {
        int mask = 1 << (3 - w);
        float ev = 0.f;
        #pragma unroll
        for (int i = 0; i < 16; ++i) {
            float p = re[i] * re[i] + im[i] * im[i];
            ev += (i & mask) ? -p : p;
        }
        Q[b * NQ_ + w] = ev;
    }
}

// ---------------------------------------------------------------------------
// K4b: out = q @ W_proj^T + b_proj via single V_WMMA_F32_16X16X4_F32 per
// wave (K=4 exactly matches the instruction's K). M=64 -> 4 tiles,
// N=512 -> 32 tiles: 128 waves. grid 16 x block 256 (8 waves/block).
// ---------------------------------------------------------------------------
__global__ void k_proj(const float* __restrict__ Q, const float* __restrict__ Wp,
                       const float* __restrict__ bp, float* __restrict__ out) {
    int wave = blockIdx.x * 8 + (threadIdx.x >> 5);
    int lane = threadIdx.x & 31;
    int mt = wave >> 5;          // 0..3
    int nt = wave & 31;          // 0..31
    int n16 = lane & 15;
    bool hi = lane >= 16;
    int kk = hi ? 2 : 0;

    int m = mt * 16 + n16;
    int n = nt * 16 + n16;
    v2f a, bm;
    a.x  = Q[m * NQ_ + kk];
    a.y  = Q[m * NQ_ + kk + 1];
    bm.x = Wp[n * NQ_ + kk];
    bm.y = Wp[n * NQ_ + kk + 1];

    v8f acc = {0.f, 0.f, 0.f, 0.f, 0.f, 0.f, 0.f, 0.f};
    acc = __builtin_amdgcn_wmma_f32_16x16x4_f32(
        false, a, false, bm, (short)0, acc, false, false);

    float bias = bp[n];
    #pragma unroll
    for (int r = 0; r < 8; ++r) {
        int row = mt * 16 + r + (hi ? 8 : 0);
        out[(size_t)row * LD_ + n] = acc[r] + bias;
    }
}

// ---------------------------------------------------------------------------
extern "C" void kernel_launch(void* const* d_in, const int* in_sizes, int n_in,
                              void* d_out, int out_size, void* d_ws, size_t ws_size,
                              hipStream_t stream) {
    const float* x      = (const float*)d_in[0];
    const float* W_red  = (const float*)d_in[1];
    const float* b_red  = (const float*)d_in[2];
    const float* weights= (const float*)d_in[3];
    const float* W_proj = (const float*)d_in[4];
    const float* b_proj = (const float*)d_in[5];
    float* out = (float*)d_out;

    float* ws = (float*)d_ws;
    float* P  = ws;                                  // [8][64][1024]
    float* XM = P + (size_t)SPLIT_ * B_ * VD_;       // [64][1024]
    float* Z  = XM + (size_t)B_ * VD_;               // [64][4]
    float* Q  = Z + B_ * NQ_;                        // [64][4]

    k_partial<<<dim3(B_, SPLIT_), 256, 0, stream>>>(x, P);
    k_reduce <<<B_, 256, 0, stream>>>(P, XM);
    k_z      <<<1, 128, 0, stream>>>(XM, W_red, b_red, Z);
    k_sim    <<<1, B_, 0, stream>>>(Z, weights, Q);
    k_proj   <<<16, 256, 0, stream>>>(Q, W_proj, b_proj, out);
}